// ProbabilisticSkipper_65979287601681
// MI455X (gfx1250) — compile-verified
//
#include <hip/hip_runtime.h>
#include <stdint.h>

#define B_ 16
#define S_ 4096
#define D_ 512
#define TILE_ 8   // output rows per gather block

// ---------------------------------------------------------------------------
// Kernel 1: per batch row, compute keep flags and a *stable* compaction of the
// kept indices via wave32 ballot + block-wide scan. One 1024-thread block
// (32 waves) per batch row; 4 iterations of 1024 over S=4096.
// ---------------------------------------------------------------------------
__global__ void ps_scan_kernel(const float* __restrict__ probs,
                               const float* __restrict__ uni,
                               int* __restrict__ idx,
                               int* __restrict__ counts) {
  const int b    = blockIdx.x;
  const int tid  = threadIdx.x;
  const int lane = tid & 31;
  const int wave = tid >> 5;

  __shared__ int waveTot[32];
  __shared__ int waveOff[32];
  __shared__ int sBase;
  if (tid == 0) sBase = 0;
  __syncthreads();

  const float* pr = probs + (size_t)b * S_;
  const float* un = uni   + (size_t)b * S_;
  int*         ib = idx   + (size_t)b * S_;

  for (int s0 = 0; s0 < S_; s0 += 1024) {
    const int  s    = s0 + tid;
    const bool keep = (un[s] >= pr[s]);

    // wave32 ballot: rank of this lane among kept lanes of its wave
    unsigned long long m = __ballot(keep);
    const int rank = __popcll(m & ((1ull << lane) - 1ull));
    const int tot  = __popcll(m);
    if (lane == 0) waveTot[wave] = tot;
    __syncthreads();

    if (tid == 0) {
      int acc = sBase;
      #pragma unroll
      for (int w = 0; w < 32; ++w) { waveOff[w] = acc; acc += waveTot[w]; }
      sBase = acc;
    }
    __syncthreads();

    if (keep) ib[waveOff[wave] + rank] = s;
  }
  __syncthreads();
  if (tid == 0) counts[b] = sBase;
}

// ---------------------------------------------------------------------------
// Kernel 2: one 128-thread block per TILE_=8 output rows of one batch.
// Phase 1: issue up to 8 async HBM->LDS row loads (b128/lane, 8 in flight).
// Phase 2: single s_wait_asynccnt 0.
// Phase 3: async LDS->HBM stores for copied rows, float4 zeros for pad rows.
// All pos<cnt conditions are block-uniform => EXEC all-ones in async regions.
// ---------------------------------------------------------------------------
__global__ void ps_gather_kernel(const float* __restrict__ x,
                                 const int* __restrict__ idx,
                                 const int* __restrict__ counts,
                                 float* __restrict__ padded,
                                 float* __restrict__ masks,
                                 int L) {
  __shared__ __align__(16) float tile[TILE_][D_];

  const int pos0 = blockIdx.x * TILE_;
  const int b    = blockIdx.y;
  const int tid  = threadIdx.x;           // 128 threads, 16B each = 2 KB/row
  const int cnt  = counts[b];

  // Phase 1: launch async loads for all rows that need a copy.
  #pragma unroll
  for (int r = 0; r < TILE_; ++r) {
    const int pos = pos0 + r;
    if (pos < cnt) {                       // block-uniform
      const int    s    = idx[b * S_ + pos];
      const float* srow = x + ((size_t)b * S_ + s) * D_;
      const unsigned           lds = (unsigned)(uintptr_t)(&tile[r][tid * 4]);
      const unsigned long long ga  = (unsigned long long)(uintptr_t)(srow + tid * 4);
      asm volatile("global_load_async_to_lds_b128 %0, %1, off"
                   :: "v"(lds), "v"(ga) : "memory");
    }
  }

  // Phase 2: all loads of this wave landed in LDS.
  asm volatile("s_wait_asynccnt 0x0" ::: "memory");

  // Phase 3: stores (async from LDS) or zero-fill; plus mask elements.
  #pragma unroll
  for (int r = 0; r < TILE_; ++r) {
    const int pos = pos0 + r;
    if (pos >= L) break;                   // block-uniform tail guard
    float* drow = padded + ((size_t)b * L + pos) * D_;
    if (pos < cnt) {                       // block-uniform
      const unsigned           lds = (unsigned)(uintptr_t)(&tile[r][tid * 4]);
      const unsigned long long gd  = (unsigned long long)(uintptr_t)(drow + tid * 4);
      asm volatile("global_store_async_from_lds_b128 %0, %1, off"
                   :: "v"(gd), "v"(lds) : "memory");
    } else {
      *(float4*)(drow + tid * 4) = make_float4(0.f, 0.f, 0.f, 0.f);
    }
  }

  if (tid < TILE_) {
    const int pos = pos0 + tid;
    if (pos < L) masks[(size_t)b * L + pos] = (pos < cnt) ? 1.0f : 0.0f;
  }

  // Drain this wave's async stores (S_ENDPGM also implies wait-idle).
  asm volatile("s_wait_asynccnt 0x0" ::: "memory");
}

extern "C" void kernel_launch(void* const* d_in, const int* in_sizes, int n_in,
                              void* d_out, int out_size, void* d_ws, size_t ws_size,
                              hipStream_t stream) {
  (void)in_sizes; (void)n_in; (void)ws_size;

  const float* x     = (const float*)d_in[0];
  const float* probs = (const float*)d_in[1];
  const float* uni   = (const float*)d_in[2];

  // out = [padded (B,L,D) | masks (B,1,L)] flat  =>  out_size = B*L*(D+1)
  const int L = out_size / (B_ * (D_ + 1));

  int* counts = (int*)d_ws;                       // 16 ints
  int* idx    = (int*)((char*)d_ws + 256);        // B*S ints = 256 KB

  float* padded = (float*)d_out;
  float* masks  = padded + (size_t)B_ * (size_t)L * (size_t)D_;

  ps_scan_kernel<<<dim3(B_), dim3(1024), 0, stream>>>(probs, uni, idx, counts);

  const int nTiles = (L + TILE_ - 1) / TILE_;
  ps_gather_kernel<<<dim3(nTiles, B_), dim3(128), 0, stream>>>(
      x, idx, counts, padded, masks, L);
}